// ConvLSTMNet_6193342841213
// MI455X (gfx1250) — compile-verified
//
#include <hip/hip_runtime.h>
#include <hip/hip_bf16.h>

typedef __attribute__((ext_vector_type(2))) float v2f;
typedef __attribute__((ext_vector_type(8))) float v8f;

#define HC   64
#define G4   256     // 4*HC gate width
#define PP   55      // H*W pixels
#define BB   32      // batch
#define TT   256     // timesteps
#define FEAT 7040    // 2*HC*PP

// ---------------------------------------------------------------------------
// Kernel 1: fused bidirectional per-pixel LSTM scan.
// grid = (110 row-blocks, 4 scans[input x dir]); block = 256 threads (8 waves).
// Each block owns 16 independent (b,p) rows for all 256 steps.
// Per step: g = h @ wh via V_WMMA_F32_16X16X4_F32 (wave w -> gate tiles 2w,2w+1),
// then gates += x-projection (IC=2, fused) + biases, elementwise LSTM update.
// wh fragments live in registers (time-invariant); h/c/g staged in LDS/regs.
// ---------------------------------------------------------------------------
__global__ void __launch_bounds__(256) convlstm_scan_kernel(
    const float* __restrict__ x1, const float* __restrict__ x2,
    const float* __restrict__ wx1, const float* __restrict__ wh1,
    const float* __restrict__ bx1, const float* __restrict__ bh1,
    const float* __restrict__ wx2, const float* __restrict__ wh2,
    const float* __restrict__ bx2, const float* __restrict__ bh2,
    float* __restrict__ feat1, float* __restrict__ feat2)
{
    __shared__ float hLDS[16][66];   // 16 rows x 64 hidden, padded (even stride, no bank conflicts)
    __shared__ float gLDS[16][G4];   // 16 rows x 256 gates
    __shared__ float xbuf[16][2];    // per-step input values for the 16 rows

    const int scan = blockIdx.y;           // 0..3
    const int inp  = scan >> 1;            // which input tensor
    const int dir  = scan & 1;             // 0 = forward cell1, 1 = reversed cell2
    const float* __restrict__ x  = inp ? x2  : x1;
    const float* __restrict__ wx = dir ? wx2 : wx1;
    const float* __restrict__ wh = dir ? wh2 : wh1;
    const float* __restrict__ bx = dir ? bx2 : bx1;
    const float* __restrict__ bh = dir ? bh2 : bh1;
    float* __restrict__ feat = inp ? feat2 : feat1;

    const int r0    = blockIdx.x * 16;     // first (b*55+p) row of this block
    const int tid   = threadIdx.x;
    const int lane  = tid & 31;
    const int wv    = tid >> 5;            // wave 0..7
    const int khalf = (lane >> 4) << 1;    // 0 (lanes 0-15) or 2 (lanes 16-31)
    const int mcol  = lane & 15;

    // ---- preload wh B-fragments: wave w owns gate tiles 2w, 2w+1 (K=64 -> 16 WMMAs each)
    v2f bfrag[2][16];
#pragma unroll
    for (int j = 0; j < 2; ++j) {
        const int col = (wv * 2 + j) * 16 + mcol;
#pragma unroll
        for (int kk = 0; kk < 16; ++kk) {
            bfrag[j][kk].x = wh[(kk * 4 + khalf    ) * G4 + col];
            bfrag[j][kk].y = wh[(kk * 4 + khalf + 1) * G4 + col];
        }
    }

    // ---- phase-2 (elementwise) ownership: thread -> channel ch, 4 rows
    const int ch = tid & 63;
    const int rg = tid >> 6;               // row group 0..3 (rows rg*4 .. rg*4+3)
    float wxc[2][4], bsum[4];
#pragma unroll
    for (int g = 0; g < 4; ++g) {
        const int n = ch + 64 * g;         // gate order i,f,o,g along 4*HC
        wxc[0][g] = wx[n];
        wxc[1][g] = wx[G4 + n];
        bsum[g]   = bx[n] + bh[n];
    }
    float creg[4] = {0.f, 0.f, 0.f, 0.f};
    float hreg[4] = {0.f, 0.f, 0.f, 0.f};

    for (int i = tid; i < 16 * 66; i += 256) (&hLDS[0][0])[i] = 0.0f;
    __syncthreads();

    for (int ts = 0; ts < TT; ++ts) {
        const int t = dir ? (TT - 1 - ts) : ts;

        // phase 0: fetch this step's x values (2 floats per row)
        if (tid < 32) {
            const int rr = tid >> 1, cc = tid & 1;
            const int row = r0 + rr;
            const int b = row / PP, p = row % PP;
            xbuf[rr][cc] = x[((b * TT + t) * 2 + cc) * PP + p];
        }
        __syncthreads();

        // phase 1: g = h @ wh  (WMMA f32 16x16x4, chained over K=64)
#pragma unroll
        for (int j = 0; j < 2; ++j) {
            const int nt = wv * 2 + j;
            v8f acc = {};
#pragma unroll
            for (int kk = 0; kk < 16; ++kk) {
                v2f a;
                a.x = hLDS[mcol][kk * 4 + khalf];
                a.y = hLDS[mcol][kk * 4 + khalf + 1];
                acc = __builtin_amdgcn_wmma_f32_16x16x4_f32(
                    false, a, false, bfrag[j][kk], (short)0, acc, false, false);
            }
            const int rbase = (lane >> 4) << 3;   // rows 0..7 or 8..15
#pragma unroll
            for (int v = 0; v < 8; ++v)
                gLDS[rbase + v][nt * 16 + mcol] = acc[v];
        }
        __syncthreads();

        // phase 2: add x-projection + biases, LSTM gate update
#pragma unroll
        for (int j = 0; j < 4; ++j) {
            const int m = rg * 4 + j;
            const float x0 = xbuf[m][0], xv = xbuf[m][1];
            const float gi = gLDS[m][ch      ] + x0 * wxc[0][0] + xv * wxc[1][0] + bsum[0];
            const float gf = gLDS[m][ch +  64] + x0 * wxc[0][1] + xv * wxc[1][1] + bsum[1];
            const float go = gLDS[m][ch + 128] + x0 * wxc[0][2] + xv * wxc[1][2] + bsum[2];
            const float gg = gLDS[m][ch + 192] + x0 * wxc[0][3] + xv * wxc[1][3] + bsum[3];
            const float si = 1.0f / (1.0f + __expf(-gi));
            const float sf = 1.0f / (1.0f + __expf(-gf));
            const float so = 1.0f / (1.0f + __expf(-go));
            const float c  = sf * creg[j] + si * tanhf(gg);
            creg[j] = c;
            const float h = so * tanhf(c);
            hreg[j] = h;
            hLDS[m][ch] = h;
        }
        __syncthreads();
    }

    // write final hidden state in torch cat((h1,h2),1).flatten() layout:
    // feat[b, (dir*64 + ch)*55 + p]
#pragma unroll
    for (int j = 0; j < 4; ++j) {
        const int row = r0 + rg * 4 + j;
        const int b = row / PP, p = row % PP;
        feat[b * FEAT + (dir * HC + ch) * PP + p] = hreg[j];
    }
}

// ---------------------------------------------------------------------------
// Kernel 2: skinny GEMM  Out(32,N) = A(32,K) @ Bw(K,N) + bias, fp32 WMMA.
// grid = (ceil(N/64), 2 M-tiles); block = 128 (4 waves, one 16x16 N-tile each).
// Memory-bound on Bw; coalesced weight reads + prefetch ahead on the stream.
// ---------------------------------------------------------------------------
__global__ void __launch_bounds__(128) fc_gemm_kernel(
    const float* __restrict__ A, const float* __restrict__ Bw,
    const float* __restrict__ bias, float* __restrict__ Out,
    int N, int K)
{
    const int lane = threadIdx.x & 31;
    const int wv   = threadIdx.x >> 5;
    const int nt   = blockIdx.x * 4 + wv;
    if (nt * 16 >= N) return;              // uniform per wave -> EXEC stays all-ones
    const int mt    = blockIdx.y;
    const int khalf = (lane >> 4) << 1;
    const int mcol  = lane & 15;
    const int ncol  = nt * 16 + mcol;
    const int ncl   = (ncol < N) ? ncol : (N - 1);   // clamp OOB columns for loads
    const int arow  = mt * 16 + mcol;

    v8f acc = {};
    const int K4 = K >> 2;                 // all K's are multiples of 4
    for (int kk = 0; kk < K4; ++kk) {
        v2f a, b;
        a.x = A[arow * K + kk * 4 + khalf];
        a.y = A[arow * K + kk * 4 + khalf + 1];
        b.x = Bw[(kk * 4 + khalf    ) * N + ncl];
        b.y = Bw[(kk * 4 + khalf + 1) * N + ncl];
        if (kk + 8 < K4)
            __builtin_prefetch(&Bw[((kk + 8) * 4 + khalf) * N + ncl], 0, 1);
        acc = __builtin_amdgcn_wmma_f32_16x16x4_f32(
            false, a, false, b, (short)0, acc, false, false);
    }

    if (ncol < N) {
        const float bb = bias[ncol];
        const int rbase = mt * 16 + ((lane >> 4) << 3);
#pragma unroll
        for (int v = 0; v < 8; ++v)
            Out[(rbase + v) * N + ncol] = acc[v] + bb;
    }
}

// ---------------------------------------------------------------------------
extern "C" void kernel_launch(void* const* d_in, const int* in_sizes, int n_in,
                              void* d_out, int out_size, void* d_ws, size_t ws_size,
                              hipStream_t stream)
{
    const float* x1  = (const float*)d_in[0];
    const float* x2  = (const float*)d_in[1];
    const float* wx1 = (const float*)d_in[2];
    const float* wh1 = (const float*)d_in[3];
    const float* bx1 = (const float*)d_in[4];
    const float* bh1 = (const float*)d_in[5];
    const float* wx2 = (const float*)d_in[6];
    const float* wh2 = (const float*)d_in[7];
    const float* bx2 = (const float*)d_in[8];
    const float* bh2 = (const float*)d_in[9];
    const float* fw2 = (const float*)d_in[10];
    const float* fb2 = (const float*)d_in[11];
    const float* fw3 = (const float*)d_in[12];
    const float* fb3 = (const float*)d_in[13];
    const float* fw4 = (const float*)d_in[14];
    const float* fb4 = (const float*)d_in[15];
    const float* fw5 = (const float*)d_in[16];
    const float* fb5 = (const float*)d_in[17];

    float* ws    = (float*)d_ws;
    float* feat1 = ws;                        // 32*7040
    float* feat2 = feat1 + BB * FEAT;         // 32*7040
    float* bufA  = feat2 + BB * FEAT;         // 32*3400
    float* bufB  = bufA  + BB * 3400;         // 32*1000
    float* bufC  = bufB  + BB * 1000;         // 32*500

    // 4 scans (2 inputs x 2 directions), 110 row-blocks each, all 256 steps fused
    convlstm_scan_kernel<<<dim3(110, 4), 256, 0, stream>>>(
        x1, x2, wx1, wh1, bx1, bh1, wx2, wh2, bx2, bh2, feat1, feat2);

    for (int inp = 0; inp < 2; ++inp) {
        const float* f = inp ? feat2 : feat1;
        float* outp = (float*)d_out + inp * BB * 50;
        // 7040 -> 3400   (213 N-tiles)
        fc_gemm_kernel<<<dim3(54, 2), 128, 0, stream>>>(f,    fw2, fb2, bufA, 3400, FEAT);
        // 3400 -> 1000   (63 N-tiles)
        fc_gemm_kernel<<<dim3(16, 2), 128, 0, stream>>>(bufA, fw3, fb3, bufB, 1000, 3400);
        // 1000 -> 500    (32 N-tiles)
        fc_gemm_kernel<<<dim3(8, 2), 128, 0, stream>>>(bufB, fw4, fb4, bufC, 500, 1000);
        // 500 -> 50      (4 N-tiles)
        fc_gemm_kernel<<<dim3(1, 2), 128, 0, stream>>>(bufC, fw5, fb5, outp, 50, 500);
    }
}